// RNN_MultiRegional_STRALM_58420145160254
// MI455X (gfx1250) — compile-verified
//
#include <hip/hip_runtime.h>
#include <math.h>

// ---------------------------------------------------------------------------
// RNN_MultiRegional_STRALM on MI455X (gfx1250)
//
// Strategy (see analysis): weights LDS-resident per WGP, fp32 WMMA
// (V_WMMA_F32_16X16X4_F32, M=16 == batch), persistent kernel with a per-step
// grid barrier; rnn_out itself is the h ring (row t-1 feeds step t).
// ---------------------------------------------------------------------------

#define HID        1024
#define NN         2048          // 2*HID
#define BATCH      16
#define TSTEPS     1000
#define NWG        64            // one WG per WGP slice; 32 output rows each
#define ROWS       32            // output neurons per WG
#define LDS_STRIDE 2052          // 2048 + 4 pad -> stride%64==4, conflict-free
#define NTHREADS   256           // 8 wave32s

typedef __attribute__((ext_vector_type(2))) float v2f;
typedef __attribute__((ext_vector_type(8))) float v8f;

__device__ __forceinline__ float clipw(float x) {
    // F.hardtanh(w, 1e-15, 1)
    return fminf(fmaxf(x, 1e-15f), 1.0f);
}

// ---------------------------------------------------------------------------
// Kernel 1: build constrained W[j][k] (row j = output neuron, col k = input),
// masked input weights u[j], and zero the barrier counters.
// ---------------------------------------------------------------------------
__global__ void build_weights_kernel(
    const float* __restrict__ s2s_w, const float* __restrict__ a2a_w,
    const float* __restrict__ a2s_w, const float* __restrict__ s2a_w,
    const float* __restrict__ inp_w,
    const float* __restrict__ s2s_mask, const float* __restrict__ s2s_fixed,
    const float* __restrict__ s2s_D,   const float* __restrict__ a2a_D,
    const float* __restrict__ a2s_mask, const float* __restrict__ d1_mask,
    float* __restrict__ W, float* __restrict__ u, int* __restrict__ counters)
{
    int idx = blockIdx.x * blockDim.x + threadIdx.x;
    if (idx < NN * NN) {
        int j = idx >> 11;        // / 2048
        int k = idx & (NN - 1);
        float w;
        if (j < HID) {
            if (k < HID) {
                // str2str = (mask*clip(w) + fixed) @ D   (D diagonal, = -I)
                int o = j * HID + k;
                w = (s2s_mask[o] * clipw(s2s_w[o]) + s2s_fixed[o]) * s2s_D[k * HID + k];
            } else {
                int kk = k - HID;
                int o = j * HID + kk;
                w = a2s_mask[o] * clipw(a2s_w[o]);      // alm2str
            }
        } else {
            int jj = j - HID;
            if (k < HID) {
                w = clipw(s2a_w[jj * HID + k]);          // str2alm
            } else {
                int kk = k - HID;
                w = clipw(a2a_w[jj * HID + kk]) * a2a_D[kk * HID + kk]; // alm2alm @ D
            }
        }
        W[idx] = w;
    }
    if (idx < NN)     u[idx] = clipw(inp_w[idx]) * d1_mask[idx];
    if (idx < TSTEPS) counters[idx] = 0;
}

// ---------------------------------------------------------------------------
// Kernel 2: persistent time-stepper. grid = NWG WGs x 256 threads.
// WG wg owns output rows [wg*32, wg*32+32); its weight slice lives in LDS.
// ---------------------------------------------------------------------------
__global__ __launch_bounds__(NTHREADS) void rnn_persistent_kernel(
    const float* __restrict__ inp,        // [B, T, 1]
    const float* __restrict__ hn,         // [1, B, 2H]
    const float* __restrict__ inhib,      // [B, T, 2H]
    const float* __restrict__ noise_hid,  // [T]
    const float* __restrict__ noise_inp,  // [T]
    const float* __restrict__ W,          // [2H, 2H]
    const float* __restrict__ u,          // [2H]
    int* __restrict__ counters,           // [T]
    float* __restrict__ out_hn,           // [B, 2H]
    float* __restrict__ out_rnn)          // [B, T, 2H]
{
    extern __shared__ float smem[];
    float* Wl  = smem;                         // ROWS * LDS_STRIDE floats
    float* red = smem + ROWS * LDS_STRIDE;     // 8 waves * 2 tiles * 256 floats

    const int tid   = threadIdx.x;
    const int wg    = blockIdx.x;
    const int j0    = wg * ROWS;
    const int wave  = tid >> 5;                // wave32
    const int lane  = tid & 31;
    const int n     = lane & 15;               // A: batch row M; B: out col N
    const int khalf = lane >> 4;               // K pair selector (lanes 16-31)

    // ---- one-time: load 32x2048 weight slice into LDS (padded rows) -------
    for (int i = tid; i < ROWS * (NN / 4); i += NTHREADS) {
        int r  = i >> 9;                       // / 512 float4s per row
        int c4 = i & 511;
        float4 v = ((const float4*)(W + (size_t)(j0 + r) * NN))[c4];
        *(float4*)(Wl + r * LDS_STRIDE + c4 * 4) = v;
    }
    __syncthreads();

    const float scale = 1.4142135624e-3f;      // sqrt(2 * 0.01 * 0.01^2)

    for (int t = 0; t < TSTEPS; ++t) {
        // h_{t-1} source: hn for t==0, else rnn_out row t-1
        const float* hbase;
        size_t strideB;
        if (t == 0) { hbase = hn;                              strideB = NN; }
        else        { hbase = out_rnn + (size_t)(t - 1) * NN;  strideB = (size_t)TSTEPS * NN; }

        if (t > 0) {
            if (tid == 0) {
                while (__hip_atomic_load(&counters[t - 1], __ATOMIC_ACQUIRE,
                                         __HIP_MEMORY_SCOPE_AGENT) < NWG)
                    __builtin_amdgcn_s_sleep(1);
            }
            __syncthreads();
            __threadfence();   // acquire: invalidate WGP caches before reading h
        }

        // ---- WMMA: S[b][j] = sum_k h[b][k] * W[j][k] -----------------------
        // wave handles K block [wave*256, wave*256+256), both 16-col tiles.
        v8f c0 = {};
        v8f c1 = {};
        const float* arow = hbase + (size_t)n * strideB;   // batch row n
        const float* brow0 = Wl + (size_t)n * LDS_STRIDE;        // tile 0, col n
        const float* brow1 = Wl + (size_t)(16 + n) * LDS_STRIDE; // tile 1, col n
        const int kb0 = wave * 256 + 2 * khalf;
        #pragma unroll 4
        for (int kc = 0; kc < 64; ++kc) {
            const int kb = kb0 + kc * 4;
            v2f a  = *(const v2f*)(arow  + kb);   // A[m][k..k+1] per lane
            v2f b0 = *(const v2f*)(brow0 + kb);   // B[k..k+1][n] = W[j][k..k+1]
            v2f b1 = *(const v2f*)(brow1 + kb);
            c0 = __builtin_amdgcn_wmma_f32_16x16x4_f32(false, a, false, b0,
                                                       (short)0, c0, false, false);
            c1 = __builtin_amdgcn_wmma_f32_16x16x4_f32(false, a, false, b1,
                                                       (short)0, c1, false, false);
        }

        // ---- spill partial 16x16 tiles to LDS for cross-wave reduction -----
        {
            float* r0 = red + (wave * 2 + 0) * 256;
            float* r1 = red + (wave * 2 + 1) * 256;
            #pragma unroll
            for (int r = 0; r < 8; ++r) {
                int e = (r + 8 * khalf) * 16 + n;   // e = b*16 + jn
                r0[e] = c0[r];
                r1[e] = c1[r];
            }
        }
        __syncthreads();

        // ---- reduce + elementwise update; 512 outputs / 256 threads --------
        const float ph = scale * noise_hid[t];
        const float pi = scale * noise_inp[t];
        #pragma unroll
        for (int gi = 0; gi < 2; ++gi) {
            int g    = tid + gi * NTHREADS;
            int tile = g >> 8;
            int e    = g & 255;
            int b    = e >> 4;
            int jn   = e & 15;
            int j    = j0 + tile * 16 + jn;
            float s = 0.0f;
            #pragma unroll
            for (int w = 0; w < 8; ++w) s += red[(w * 2 + tile) * 256 + e];
            float hprev = hbase[(size_t)b * strideB + j];
            float it    = (inp[b * TSTEPS + t] + pi) * u[j];
            float ih    = inhib[((size_t)b * TSTEPS + t) * NN + j];
            float pre   = -hprev + s + it + ih;
            float hnew  = fmaxf(hprev + 0.01f * pre + ph, 0.0f);
            out_rnn[((size_t)b * TSTEPS + t) * NN + j] = hnew;
        }

        // ---- publish h_t ---------------------------------------------------
        __threadfence();       // release: make our stores device-visible
        __syncthreads();       // also protects `red` reuse next iteration
        if (tid == 0)
            __hip_atomic_fetch_add(&counters[t], 1, __ATOMIC_RELEASE,
                                   __HIP_MEMORY_SCOPE_AGENT);
    }

    // ---- hn_last = rnn_out[:, T-1, :]  (our own slice) ---------------------
    __threadfence();
    for (int g = tid; g < ROWS * BATCH; g += NTHREADS) {
        int b = g >> 5;            // / ROWS
        int jr = g & (ROWS - 1);
        int j = j0 + jr;
        out_hn[(size_t)b * NN + j] =
            out_rnn[((size_t)b * TSTEPS + (TSTEPS - 1)) * NN + j];
    }
}

// ---------------------------------------------------------------------------
extern "C" void kernel_launch(void* const* d_in, const int* in_sizes, int n_in,
                              void* d_out, int out_size, void* d_ws, size_t ws_size,
                              hipStream_t stream) {
    const float* inp      = (const float*)d_in[0];
    const float* hn       = (const float*)d_in[1];
    const float* inhib    = (const float*)d_in[2];
    const float* s2s_w    = (const float*)d_in[3];
    const float* a2a_w    = (const float*)d_in[4];
    const float* a2s_w    = (const float*)d_in[5];
    const float* s2a_w    = (const float*)d_in[6];
    const float* inp_w    = (const float*)d_in[7];
    const float* s2s_mask = (const float*)d_in[8];
    const float* s2s_fix  = (const float*)d_in[9];
    const float* s2s_D    = (const float*)d_in[10];
    const float* a2a_D    = (const float*)d_in[11];
    const float* a2s_mask = (const float*)d_in[12];
    const float* d1_mask  = (const float*)d_in[13];
    const float* nh       = (const float*)d_in[14];
    const float* ni       = (const float*)d_in[15];

    // workspace layout: W [2H*2H] f32 | u [2H] f32 | counters [T] i32
    float* W        = (float*)d_ws;
    float* u        = W + (size_t)NN * NN;
    int*   counters = (int*)(u + NN);

    // output layout: hn_last [B*2H] then rnn_out [B*T*2H]
    float* out_hn  = (float*)d_out;
    float* out_rnn = out_hn + (size_t)BATCH * NN;

    build_weights_kernel<<<(NN * NN + 255) / 256, 256, 0, stream>>>(
        s2s_w, a2a_w, a2s_w, s2a_w, inp_w,
        s2s_mask, s2s_fix, s2s_D, a2a_D, a2s_mask, d1_mask,
        W, u, counters);

    const size_t lds_bytes = (size_t)(ROWS * LDS_STRIDE + 8 * 2 * 256) * sizeof(float);
    (void)hipFuncSetAttribute((const void*)rnn_persistent_kernel,
                              hipFuncAttributeMaxDynamicSharedMemorySize,
                              (int)lds_bytes);

    rnn_persistent_kernel<<<NWG, NTHREADS, lds_bytes, stream>>>(
        inp, hn, inhib, nh, ni, W, u, counters, out_hn, out_rnn);

    (void)in_sizes; (void)n_in; (void)out_size; (void)ws_size;
}